// TA_GNN_51702816309815
// MI455X (gfx1250) — compile-verified
//
#include <hip/hip_runtime.h>
#include <hip/hip_bf16.h>
#include <math.h>

#define D      100
#define NN     49152
#define EE     98304
#define BB     4096
#define NNODE  50000

typedef __attribute__((ext_vector_type(2))) float v2f;
typedef __attribute__((ext_vector_type(8))) float v8f;

// ---------------- init ----------------
__global__ __launch_bounds__(256) void k_init(float* __restrict__ agg,
                                              float* __restrict__ Ss,
                                              float* __restrict__ Sg,
                                              float* __restrict__ denom,
                                              float* __restrict__ mx) {
    int i = blockIdx.x * blockDim.x + threadIdx.x;
    if (i < NN * D) agg[i] = 0.f;
    if (i < BB * D) { Ss[i] = 0.f; Sg[i] = 0.f; }
    if (i < BB)     { denom[i] = 0.f; mx[i] = -3.0e38f; }
}

// ---------------- embedding gather + m = h0 @ conv_w ----------------
__global__ __launch_bounds__(128) void k_embed_mm(const int* __restrict__ x,
                                                  const float* __restrict__ emb,
                                                  const float* __restrict__ conv_w,
                                                  float* __restrict__ h0,
                                                  float* __restrict__ m) {
    __shared__ float sh[D];
    int node = blockIdx.x;
    int t = threadIdx.x;
    int id = x[node] - 1;
    if (t < D) {
        float v = emb[(size_t)id * D + t];
        sh[t] = v;
        h0[(size_t)node * D + t] = v;
    }
    __syncthreads();
    if (t < D) {
        float acc = 0.f;
        #pragma unroll 4
        for (int k = 0; k < D; ++k) acc += sh[k] * conv_w[k * D + t];
        m[(size_t)node * D + t] = acc;
    }
}

// ---------------- edge scatter: agg[dst] += m[src] ----------------
__global__ __launch_bounds__(128) void k_scatter(const int* __restrict__ ei,
                                                 const float* __restrict__ m,
                                                 float* __restrict__ agg) {
    int e = blockIdx.x;
    int t = threadIdx.x;
    if (t >= D) return;
    int src = ei[e];
    int dst = ei[EE + e];
    atomicAdd(&agg[(size_t)dst * D + t], m[(size_t)src * D + t]);
}

// ---------------- GRU cell update via WMMA ----------------
// Per block: 16 nodes. gi = agg @ w_ih^T, gh = h0 @ w_hh^T  (N=300, K=100),
// computed as 2 x 19 tiles of 16x16 by 8 waves, then fused GRU elementwise.
__global__ __launch_bounds__(256) void k_gru_wmma(const float* __restrict__ agg,
                                                  const float* __restrict__ h0,
                                                  const float* __restrict__ w_ih,
                                                  const float* __restrict__ w_hh,
                                                  const float* __restrict__ b_ih,
                                                  const float* __restrict__ b_hh,
                                                  float* __restrict__ xh) {
    __shared__ float sIn[2][16][102];   // [0]=agg rows, [1]=h0 rows
    __shared__ float sG[2][16][308];    // [0]=gi, [1]=gh (cols 300..307 = pad)
    const int wave = threadIdx.x >> 5;
    const int lane = threadIdx.x & 31;
    const int nodeBase = blockIdx.x * 16;

    for (int idx = threadIdx.x; idx < 16 * D; idx += 256) {
        int r = idx / D, c = idx % D;
        sIn[0][r][c] = agg[(size_t)(nodeBase + r) * D + c];
        sIn[1][r][c] = h0[(size_t)(nodeBase + r) * D + c];
    }
    __syncthreads();

    const int row   = lane & 15;
    const int kHalf = (lane >> 4) * 2;   // A: lanes 0-15 K={0,1}, lanes 16-31 K={2,3}
    const int mOff  = (lane >> 4) * 8;   // D: lanes 0-15 rows v, lanes 16-31 rows v+8

    // 38 tiles: tile/19 selects gemm (0=gi via w_ih, 1=gh via w_hh), tile%19 = col tile
    for (int tile = wave; tile < 38; tile += 8) {
        int g  = tile / 19;
        int ct = tile - g * 19;
        int col = ct * 16 + (lane & 15);
        if (col > 299) col = 299;        // clamp: pad cols give garbage, never read
        const float* wRow = (g ? w_hh : w_ih) + (size_t)col * D;

        v8f acc = {0.f, 0.f, 0.f, 0.f, 0.f, 0.f, 0.f, 0.f};
        #pragma unroll
        for (int k0 = 0; k0 < D; k0 += 4) {
            v2f a, b;
            a.x = sIn[g][row][k0 + kHalf];
            a.y = sIn[g][row][k0 + kHalf + 1];
            b = *(const v2f*)(wRow + k0 + kHalf);   // B[k][n] = w[n][k], contiguous
            acc = __builtin_amdgcn_wmma_f32_16x16x4_f32(false, a, false, b,
                                                        (short)0, acc, false, false);
        }
        int c = ct * 16 + (lane & 15);
        #pragma unroll
        for (int v = 0; v < 8; ++v) sG[g][mOff + v][c] = acc[v];
    }
    __syncthreads();

    for (int idx = threadIdx.x; idx < 16 * D; idx += 256) {
        int nn = idx / D, t = idx % D;
        float gi_r = sG[0][nn][t]         + b_ih[t];
        float gi_z = sG[0][nn][D + t]     + b_ih[D + t];
        float gi_n = sG[0][nn][2 * D + t] + b_ih[2 * D + t];
        float gh_r = sG[1][nn][t]         + b_hh[t];
        float gh_z = sG[1][nn][D + t]     + b_hh[D + t];
        float gh_n = sG[1][nn][2 * D + t] + b_hh[2 * D + t];
        float r = 1.f / (1.f + __expf(-(gi_r + gh_r)));
        float z = 1.f / (1.f + __expf(-(gi_z + gh_z)));
        float n = tanhf(gi_n + r * gh_n);
        float v = (1.f - z) * n + z * sIn[1][nn][t];
        xh[(size_t)(nodeBase + nn) * D + t] = v > 0.f ? v : 0.f;
    }
}

// ---------------- last node per session -> v_n ----------------
__global__ __launch_bounds__(256) void k_lastnode(const int* __restrict__ batch,
                                                  const float* __restrict__ xh,
                                                  float* __restrict__ vn) {
    int i = blockIdx.x * blockDim.x + threadIdx.x;
    if (i >= NN) return;
    if (i == NN - 1 || batch[i + 1] != batch[i]) {
        int s = batch[i];
        for (int k = 0; k < D; ++k) vn[(size_t)s * D + k] = xh[(size_t)i * D + k];
    }
}

__device__ inline void atomicMaxFloat(float* addr, float val) {
    if (val >= 0.0f) atomicMax((int*)addr, __float_as_int(val));
    else             atomicMin((unsigned int*)addr, (unsigned int)__float_as_int(val));
}

// ---------------- logits + segment max (wave per node) ----------------
__global__ __launch_bounds__(256) void k_logits(const int* __restrict__ batch,
                                                const float* __restrict__ xh,
                                                const float* __restrict__ vn,
                                                float* __restrict__ logits,
                                                float* __restrict__ mx) {
    int node = blockIdx.x * 8 + (threadIdx.x >> 5);
    int lane = threadIdx.x & 31;
    if (node >= NN) return;
    int b = batch[node];
    float s = 0.f;
    for (int k = lane; k < D; k += 32)
        s += xh[(size_t)node * D + k] * vn[(size_t)b * D + k];
    for (int off = 16; off > 0; off >>= 1) s += __shfl_down(s, off, 32);
    if (lane == 0) {
        logits[node] = s;
        atomicMaxFloat(&mx[b], s);
    }
}

// ---------------- p = exp(logits - mx), denom ----------------
__global__ __launch_bounds__(256) void k_softmax_p(const int* __restrict__ batch,
                                                   const float* __restrict__ logits,
                                                   const float* __restrict__ mx,
                                                   float* __restrict__ p,
                                                   float* __restrict__ denom) {
    int i = blockIdx.x * blockDim.x + threadIdx.x;
    if (i >= NN) return;
    int b = batch[i];
    float v = __expf(logits[i] - mx[b]);
    p[i] = v;
    atomicAdd(&denom[b], v);
}

// ---------------- attention: S_s and S_g accumulation ----------------
__global__ __launch_bounds__(128) void k_attn(const int* __restrict__ batch,
                                              const float* __restrict__ xh,
                                              const float* __restrict__ vn,
                                              const float* __restrict__ p,
                                              const float* __restrict__ denom,
                                              const float* __restrict__ w2w,
                                              const float* __restrict__ w2b,
                                              const float* __restrict__ w3w,
                                              const float* __restrict__ w3b,
                                              const float* __restrict__ qw,
                                              const float* __restrict__ qb,
                                              float* __restrict__ Ss,
                                              float* __restrict__ Sg) {
    __shared__ float s_x[D];
    __shared__ float s_v[D];
    __shared__ float s_c[128];
    int i = blockIdx.x;
    int t = threadIdx.x;
    int b = batch[i];
    if (t < D) {
        s_x[t] = xh[(size_t)i * D + t];
        s_v[t] = vn[(size_t)b * D + t];
    }
    __syncthreads();
    float contrib = 0.f;
    if (t < D) {
        float g = w2b[t] + w3b[t];
        for (int k = 0; k < D; ++k)
            g += w2w[t * D + k] * s_x[k] + w3w[t * D + k] * s_v[k];
        contrib = qw[t] / (1.f + __expf(-g));
    }
    s_c[t] = contrib;
    __syncthreads();
    for (int s = 64; s > 0; s >>= 1) {
        if (t < s) s_c[t] += s_c[t + s];
        __syncthreads();
    }
    float alpha = s_c[0] + qb[0];
    float score = p[i] / denom[b];
    if (t < D) {
        atomicAdd(&Ss[(size_t)b * D + t], score * s_x[t]);
        atomicAdd(&Sg[(size_t)b * D + t], alpha * s_x[t]);
    }
}

// ---------------- S = concat(S_s, S_g, v_n) @ w4^T + b4 ----------------
__global__ __launch_bounds__(128) void k_wcat(const float* __restrict__ Ss,
                                              const float* __restrict__ Sg,
                                              const float* __restrict__ vn,
                                              const float* __restrict__ w4w,
                                              const float* __restrict__ w4b,
                                              float* __restrict__ S) {
    __shared__ float cat[3 * D];
    int b = blockIdx.x;
    int t = threadIdx.x;
    if (t < D) {
        cat[t]         = Ss[(size_t)b * D + t];
        cat[D + t]     = Sg[(size_t)b * D + t];
        cat[2 * D + t] = vn[(size_t)b * D + t];
    }
    __syncthreads();
    if (t < D) {
        float acc = w4b[t];
        for (int j = 0; j < 3 * D; ++j) acc += cat[j] * w4w[t * 3 * D + j];
        S[(size_t)b * D + t] = acc;
    }
}

// ---------------- y_hat = S @ emb^T via V_WMMA_F32_16X16X4_F32 ----------------
// Block: 256 threads = 8 waves. Block tile: 32(M) x 128(N). Each wave computes
// two 16x16 M-tiles sharing one B load stream. Output staged in LDS for fully
// coalesced stores (store-bandwidth-bound kernel: 819 MB output).
#define TM 32
#define TN 128
__global__ __launch_bounds__(256) void k_yhat(const float* __restrict__ S,
                                              const float* __restrict__ emb,
                                              float* __restrict__ out) {
    __shared__ float sA[TM][102];
    __shared__ float sO[TM][TN + 4];
    const int wave = threadIdx.x >> 5;
    const int lane = threadIdx.x & 31;
    const int mBase = blockIdx.y * TM;
    const int nBase = blockIdx.x * TN;

    for (int idx = threadIdx.x; idx < TM * D; idx += 256) {
        int r = idx / D, c = idx % D;
        sA[r][c] = S[(size_t)(mBase + r) * D + c];
    }
    __syncthreads();

    const int row   = lane & 15;
    const int kHalf = (lane >> 4) * 2;
    int col = nBase + wave * 16 + (lane & 15);
    if (col >= NNODE) col = NNODE - 1;          // clamp: OOB lanes compute garbage,
    const float* embRow = emb + (size_t)col * D; // stores are guarded below
    __builtin_prefetch(embRow, 0, 3);

    v8f acc0 = {0.f, 0.f, 0.f, 0.f, 0.f, 0.f, 0.f, 0.f};
    v8f acc1 = {0.f, 0.f, 0.f, 0.f, 0.f, 0.f, 0.f, 0.f};
    #pragma unroll
    for (int k0 = 0; k0 < D; k0 += 4) {
        v2f b = *(const v2f*)(embRow + k0 + kHalf);  // B[k][n] = emb[n][k]
        v2f a0, a1;
        a0.x = sA[row][k0 + kHalf];
        a0.y = sA[row][k0 + kHalf + 1];
        a1.x = sA[16 + row][k0 + kHalf];
        a1.y = sA[16 + row][k0 + kHalf + 1];
        acc0 = __builtin_amdgcn_wmma_f32_16x16x4_f32(false, a0, false, b,
                                                     (short)0, acc0, false, false);
        acc1 = __builtin_amdgcn_wmma_f32_16x16x4_f32(false, a1, false, b,
                                                     (short)0, acc1, false, false);
    }

    // D layout: VGPR v -> M = v (lanes 0-15) / v+8 (lanes 16-31), N = lane&15
    {
        int mOff = (lane >> 4) * 8;
        int c = wave * 16 + (lane & 15);
        #pragma unroll
        for (int v = 0; v < 8; ++v) {
            sO[mOff + v][c]      = acc0[v];
            sO[16 + mOff + v][c] = acc1[v];
        }
    }
    __syncthreads();

    for (int idx = threadIdx.x; idx < TM * TN; idx += 256) {
        int r = idx / TN, cc = idx % TN;
        int gcol = nBase + cc;
        if (gcol < NNODE)
            out[(size_t)(mBase + r) * NNODE + gcol] = sO[r][cc];
    }
}

extern "C" void kernel_launch(void* const* d_in, const int* in_sizes, int n_in,
                              void* d_out, int out_size, void* d_ws, size_t ws_size,
                              hipStream_t stream) {
    const int*   x      = (const int*)d_in[0];
    const int*   ei     = (const int*)d_in[1];
    const int*   batch  = (const int*)d_in[2];
    const float* emb    = (const float*)d_in[4];
    const float* conv_w = (const float*)d_in[5];
    const float* w_ih   = (const float*)d_in[6];
    const float* w_hh   = (const float*)d_in[7];
    const float* b_ih   = (const float*)d_in[8];
    const float* b_hh   = (const float*)d_in[9];
    const float* qw     = (const float*)d_in[10];
    const float* qb     = (const float*)d_in[11];
    const float* w2w    = (const float*)d_in[12];
    const float* w2b    = (const float*)d_in[13];
    const float* w3w    = (const float*)d_in[14];
    const float* w3b    = (const float*)d_in[15];
    const float* w4w    = (const float*)d_in[16];
    const float* w4b    = (const float*)d_in[17];
    float* out = (float*)d_out;

    float* ws     = (float*)d_ws;
    float* h0     = ws;
    float* m      = h0 + (size_t)NN * D;
    float* agg    = m + (size_t)NN * D;
    float* xh     = agg + (size_t)NN * D;
    float* vn     = xh + (size_t)NN * D;
    float* logits = vn + (size_t)BB * D;
    float* pbuf   = logits + NN;
    float* mx     = pbuf + NN;
    float* denom  = mx + BB;
    float* Ss     = denom + BB;
    float* Sg     = Ss + (size_t)BB * D;
    float* Sfin   = Sg + (size_t)BB * D;

    k_init<<<(NN * D + 255) / 256, 256, 0, stream>>>(agg, Ss, Sg, denom, mx);
    k_embed_mm<<<NN, 128, 0, stream>>>(x, emb, conv_w, h0, m);
    k_scatter<<<EE, 128, 0, stream>>>(ei, m, agg);
    k_gru_wmma<<<NN / 16, 256, 0, stream>>>(agg, h0, w_ih, w_hh, b_ih, b_hh, xh);
    k_lastnode<<<(NN + 255) / 256, 256, 0, stream>>>(batch, xh, vn);
    k_logits<<<NN / 8, 256, 0, stream>>>(batch, xh, vn, logits, mx);
    k_softmax_p<<<(NN + 255) / 256, 256, 0, stream>>>(batch, logits, mx, pbuf, denom);
    k_attn<<<NN, 128, 0, stream>>>(batch, xh, vn, pbuf, denom,
                                   w2w, w2b, w3w, w3b, qw, qb, Ss, Sg);
    k_wcat<<<BB, 128, 0, stream>>>(Ss, Sg, vn, w4w, w4b, Sfin);

    dim3 gy((NNODE + TN - 1) / TN, BB / TM);
    k_yhat<<<gy, 256, 0, stream>>>(Sfin, emb, out);
}